// Projector_30983894073469
// MI455X (gfx1250) — compile-verified
//
#include <hip/hip_runtime.h>

// ---------------------------------------------------------------------------
// Parallel-beam CT forward projection for MI455X (gfx1250).
//
// vol:  [1, 32, 256, 256] f32   phis: [180] f32 (degrees)
// out:  [1, 180, 32, 256] f32
//
// Strategy: one workgroup owns one z-slice (256 KB, staged in 320 KB LDS via
// CDNA5 async global->LDS copies, ASYNCcnt-tracked) and a group of 4 view
// angles. All bilinear gathers then hit LDS (bank-skewed rows), independent
// of ray direction. Volume (8 MB) is L2-resident, so fills are cheap.
// ---------------------------------------------------------------------------

#define NX    256
#define NY    256
#define NZ    32
#define NT    384
#define NANG  180
#define LSTR  260          // padded LDS row stride (floats): skew 4 mod 64 banks
#define ANG_PER_BLK 4

typedef int v4i __attribute__((__vector_size__(16)));
typedef __attribute__((address_space(1))) v4i gv4i;   // global (AS1) int4
typedef __attribute__((address_space(3))) v4i lv4i;   // LDS (AS3) int4
typedef __attribute__((address_space(3))) float lds_f32;

__device__ __forceinline__ int clampi(int v, int lo, int hi) {
    return v < lo ? lo : (v > hi ? hi : v);
}

extern "C" __global__ __launch_bounds__(256, 1)
void ct_fwdproj_kernel(const float* __restrict__ vol,
                       const float* __restrict__ phis,
                       float* __restrict__ out)
{
    extern __shared__ float sm[];                 // [NY][LSTR]
    const int z    = blockIdx.x;                  // 0..31
    const int agrp = blockIdx.y;                  // 0..44
    const int tid  = threadIdx.x;                 // 0..255 (detector col; slice row for fill)

    // ---- Stage slice z into LDS: row `tid` (1 KB) as 64 async 16B copies ----
    const float* grow = vol + (size_t)z * (NX * NY) + (size_t)tid * NX;
    lds_f32*     lrow = (lds_f32*)&sm[tid * LSTR];

    #pragma unroll
    for (int k = 0; k < NX / 4; ++k) {
#if __has_builtin(__builtin_amdgcn_global_load_async_to_lds_b128)
        __builtin_amdgcn_global_load_async_to_lds_b128(
            (gv4i*)(grow + 4 * k), (lv4i*)(lrow + 4 * k), 0, 0);
#else
        asm volatile("global_load_async_to_lds_b128 %0, %1, off"
                     :
                     : "v"((unsigned)(unsigned long long)(lrow + 4 * k)),
                       "v"(grow + 4 * k)
                     : "memory");
#endif
    }
#if __has_builtin(__builtin_amdgcn_s_wait_asynccnt)
    __builtin_amdgcn_s_wait_asynccnt(0);
#else
    asm volatile("s_wait_asynccnt 0" ::: "memory");
#endif
    __syncthreads();

    // ---- Ray march: 4 angles x 384 t-samples, column = tid -----------------
    const float uu = (float)tid - 127.5f;         // detector lateral offset

    #pragma unroll
    for (int a = 0; a < ANG_PER_BLK; ++a) {
        const int angle = agrp * ANG_PER_BLK + a;
        const float phi = phis[angle] * 0.017453292519943295f;  // deg -> rad
        const float c = cosf(phi);
        const float s = sinf(phi);

        // world->voxel: ix = (-s*u + c*t) + 127.5 ; iy = (c*u + s*t) + 127.5
        const float bx = fmaf(uu, -s, 127.5f);
        const float by = fmaf(uu,  c, 127.5f);

        float acc = 0.0f;

        #pragma unroll 4
        for (int i = 0; i < NT; ++i) {
            const float tv = (float)i - 191.5f;
            const float x = fmaf(tv, c, bx);
            const float y = fmaf(tv, s, by);

            const float xf = floorf(x), yf = floorf(y);
            const float fx = x - xf,    fy = y - yf;
            const int ix0 = (int)xf, iy0 = (int)yf;
            const int ix1 = ix0 + 1, iy1 = iy0 + 1;

            // zero-weight out-of-range taps (matches reference valid_clip)
            const float wx0 = (ix0 >= 0 && ix0 < NX) ? (1.0f - fx) : 0.0f;
            const float wx1 = (ix1 >= 0 && ix1 < NX) ? fx          : 0.0f;
            const float wy0 = (iy0 >= 0 && iy0 < NY) ? (1.0f - fy) : 0.0f;
            const float wy1 = (iy1 >= 0 && iy1 < NY) ? fy          : 0.0f;

            const int jx0 = clampi(ix0, 0, NX - 1), jx1 = clampi(ix1, 0, NX - 1);
            const int jy0 = clampi(iy0, 0, NY - 1), jy1 = clampi(iy1, 0, NY - 1);

            const float v00 = sm[jy0 * LSTR + jx0];
            const float v01 = sm[jy0 * LSTR + jx1];
            const float v10 = sm[jy1 * LSTR + jx0];
            const float v11 = sm[jy1 * LSTR + jx1];

            acc += fmaf(wy0, fmaf(wx0, v00, wx1 * v01),
                        wy1 * fmaf(wx0, v10, wx1 * v11));
        }

        // out[0, angle, z, col], col == tid  (coalesced)
        out[((size_t)angle * NZ + z) * NX + tid] = acc;  // * VOXEL_W (== 1.0)
    }
}

extern "C" void kernel_launch(void* const* d_in, const int* in_sizes, int n_in,
                              void* d_out, int out_size, void* d_ws, size_t ws_size,
                              hipStream_t stream) {
    const float* vol  = (const float*)d_in[0];   // [1,32,256,256] f32
    const float* phis = (const float*)d_in[1];   // [180] f32
    float* out = (float*)d_out;                  // [1,180,32,256] f32

    dim3 grid(NZ, NANG / ANG_PER_BLK);           // (32, 45)
    dim3 block(256);
    size_t shbytes = (size_t)NY * LSTR * sizeof(float);  // 266,240 B < 320 KB
    ct_fwdproj_kernel<<<grid, block, shbytes, stream>>>(vol, phis, out);
}